// TopKAbsolutes1D_33938831573313
// MI455X (gfx1250) — compile-verified
//
#include <hip/hip_runtime.h>
#include <hip/hip_bf16.h>
#include <stdint.h>

// Row-wise top-K (K=512) by |x| over W=16384, signed values kept, zeros elsewhere.
// One 512-thread block per row. Row lives in LDS (padded layout), exact threshold
// via 4-pass byte radix select, stable tie-break by index (matches lax.top_k).
// Data movement uses CDNA5 async global<->LDS instructions (ASYNCcnt path).

#define NTHREADS 512
#define ROWLEN   16384
#define KSEL     512
#define NWAVES   (NTHREADS / 32)

// padded LDS slot: one extra dword per 32 elements -> conflict-free chunked access
__device__ __forceinline__ int dslot(int i) { return i + (i >> 5); }

__device__ __forceinline__ void async_load_b128(uint32_t lds_off, const void* gaddr) {
    asm volatile("global_load_async_to_lds_b128 %0, %1, off"
                 :: "v"(lds_off), "v"(gaddr) : "memory");
}

__device__ __forceinline__ void async_store_b128(void* gaddr, uint32_t lds_off) {
    asm volatile("global_store_async_from_lds_b128 %0, %1, off"
                 :: "v"(gaddr), "v"(lds_off) : "memory");
}

__device__ __forceinline__ void wait_async0() {
    asm volatile("s_wait_asynccnt 0" ::: "memory");
}

__global__ __launch_bounds__(NTHREADS) void
TopKAbsolutes1D_kernel(const float* __restrict__ x, float* __restrict__ out) {
    extern __shared__ uint32_t smem[];
    uint32_t* sData = smem;                       // (16384 + 512) dwords, padded row
    uint32_t* sHist = sData + (ROWLEN + ROWLEN / 32); // NWAVES * 256
    uint32_t* sComb = sHist + NWAVES * 256;       // 256
    uint32_t* sScan = sComb + 256;                // 512
    uint32_t* sMisc = sScan + NTHREADS;           // [0]=prefix, [1]=remaining

    const int tid  = threadIdx.x;
    const int wave = tid >> 5;
    const size_t rowBase = (size_t)blockIdx.x * ROWLEN;
    const float* rowIn  = x   + rowBase;
    float*       rowOut = out + rowBase;

    // ---- Phase 1: async copy row into LDS (padded layout) -------------------
    // 4096 float4 transfers; groups of 4 dwords never straddle a pad boundary.
    #pragma unroll
    for (int it = 0; it < ROWLEN / (NTHREADS * 4); ++it) {
        int i = (tid + it * NTHREADS) * 4;
        uint32_t ldsOff = (uint32_t)(uintptr_t)(sData + dslot(i));
        async_load_b128(ldsOff, (const void*)(rowIn + i));
    }
    wait_async0();
    __syncthreads();

    // ---- Phase 2: 4-pass radix select on |x| bit patterns -------------------
    uint32_t pref = 0;
    int remaining = KSEL;

    for (int p = 3; p >= 0; --p) {
        // zero per-wave histograms
        for (int z = tid; z < NWAVES * 256; z += NTHREADS) sHist[z] = 0u;
        __syncthreads();

        const uint32_t maskHi = (p == 3) ? 0u : (0xFFFFFFFFu << ((p + 1) * 8));
        const uint32_t prefHi = pref & maskHi;
        uint32_t* myHist = sHist + (wave << 8);

        #pragma unroll 4
        for (int k = 0; k < ROWLEN / NTHREADS; ++k) {
            int i = tid + k * NTHREADS;
            uint32_t bits = sData[dslot(i)] & 0x7FFFFFFFu;
            if ((bits & maskHi) == prefHi) {
                uint32_t bin = (bits >> (p * 8)) & 0xFFu;
                atomicAdd(&myHist[bin], 1u);
            }
        }
        __syncthreads();

        // combine per-wave histograms
        if (tid < 256) {
            uint32_t s = 0;
            #pragma unroll
            for (int w = 0; w < NWAVES; ++w) s += sHist[(w << 8) + tid];
            sComb[tid] = s;
        }
        __syncthreads();

        // inclusive suffix-sum over 256 bins (count of elements with byte >= b)
        for (int off = 1; off < 256; off <<= 1) {
            uint32_t v = 0;
            if (tid < 256) {
                v = sComb[tid];
                if (tid + off < 256) v += sComb[tid + off];
            }
            __syncthreads();
            if (tid < 256) sComb[tid] = v;
            __syncthreads();
        }

        // pick the byte: ge[b] >= remaining > ge[b+1]
        if (tid < 256) {
            int ge     = (int)sComb[tid];
            int geNext = (tid == 255) ? 0 : (int)sComb[tid + 1];
            if (ge >= remaining && geNext < remaining) {
                sMisc[0] = pref | ((uint32_t)tid << (p * 8));
                sMisc[1] = (uint32_t)(remaining - geNext);
            }
        }
        __syncthreads();
        pref      = sMisc[0];
        remaining = (int)sMisc[1];
        __syncthreads();
    }

    const uint32_t T = pref;           // exact K-th largest |x| bit pattern
    const int remEq  = remaining;      // number of ==T elements to keep (lowest index first)

    // ---- Phase 3: stable tie ranking (contiguous 32-element chunks) ---------
    const int base = tid * 32;
    int cntEq = 0;
    #pragma unroll 8
    for (int j = 0; j < 32; ++j) {
        uint32_t bits = sData[dslot(base + j)] & 0x7FFFFFFFu;
        cntEq += (bits == T) ? 1 : 0;
    }
    sScan[tid] = (uint32_t)cntEq;
    __syncthreads();
    for (int off = 1; off < NTHREADS; off <<= 1) {
        uint32_t v = sScan[tid];
        if (tid >= off) v += sScan[tid - off];
        __syncthreads();
        sScan[tid] = v;
        __syncthreads();
    }
    int rank = (tid == 0) ? 0 : (int)sScan[tid - 1];  // equals before my chunk

    // overwrite LDS row in place with the output values
    #pragma unroll 8
    for (int j = 0; j < 32; ++j) {
        int di = dslot(base + j);
        uint32_t raw  = sData[di];
        uint32_t bits = raw & 0x7FFFFFFFu;
        bool keep;
        if (bits > T)       keep = true;
        else if (bits == T) { keep = (rank < remEq); ++rank; }
        else                keep = false;
        sData[di] = keep ? raw : 0u;
    }
    __syncthreads();

    // ---- Phase 4: async store row LDS -> global -----------------------------
    #pragma unroll
    for (int it = 0; it < ROWLEN / (NTHREADS * 4); ++it) {
        int i = (tid + it * NTHREADS) * 4;
        uint32_t ldsOff = (uint32_t)(uintptr_t)(sData + dslot(i));
        async_store_b128((void*)(rowOut + i), ldsOff);
    }
    wait_async0();
}

extern "C" void kernel_launch(void* const* d_in, const int* in_sizes, int n_in,
                              void* d_out, int out_size, void* d_ws, size_t ws_size,
                              hipStream_t stream) {
    (void)n_in; (void)d_ws; (void)ws_size; (void)out_size;
    const float* x = (const float*)d_in[0];
    float* out = (float*)d_out;
    int rows = in_sizes[0] / ROWLEN;   // 32*64 = 2048

    size_t shmem = (size_t)((ROWLEN + ROWLEN / 32) + NWAVES * 256 + 256 + NTHREADS + 4)
                   * sizeof(uint32_t); // ~87 KB of the 320 KB WGP LDS
    TopKAbsolutes1D_kernel<<<dim3(rows), dim3(NTHREADS), shmem, stream>>>(x, out);
}